// GNN_20504173871436
// MI455X (gfx1250) — compile-verified
//
#include <hip/hip_runtime.h>
#include <hip/hip_bf16.h>

typedef float v2f __attribute__((ext_vector_type(2)));
typedef float v8f __attribute__((ext_vector_type(8)));

// ---------------------------------------------------------------- utilities
__global__ void zero_floats(float* __restrict__ p, int n) {
    int i = blockIdx.x * blockDim.x + threadIdx.x;
    if (i < n) p[i] = 0.0f;
}

// out[i] = (1+eps)*in[i], float4-vectorized (n4 = count/4)
__global__ void scale_copy4(const float* __restrict__ in, float* __restrict__ out,
                            const float* __restrict__ eps, long long n4) {
    long long i = (long long)blockIdx.x * blockDim.x + threadIdx.x;
    long long stride = (long long)gridDim.x * blockDim.x;
    const float s = 1.0f + eps[0];
    for (; i < n4; i += stride) {
        float4 v = ((const float4*)in)[i];
        v.x *= s; v.y *= s; v.z *= s; v.w *= s;
        ((float4*)out)[i] = v;
    }
}

// ------------------------------------------------------- edge scatter (GIN agg)
// Layer 1: 128 features, one wave per edge, 4 floats per lane.
__global__ void scatter_edges_128(const float* __restrict__ x, float* __restrict__ acc,
                                  const int* __restrict__ src, const int* __restrict__ dst,
                                  long long E) {
    long long gid = (long long)blockIdx.x * blockDim.x + threadIdx.x;
    long long stride = (long long)gridDim.x * blockDim.x;
    for (long long w = gid; w < E * 32; w += stride) {
        long long e = w >> 5;
        int lane = (int)(w & 31);
        int s = src[e], d = dst[e];
        const float4 v = *(const float4*)(x + (size_t)s * 128 + lane * 4);
        float* p = acc + (size_t)d * 128 + lane * 4;
        atomicAdd(p + 0, v.x);
        atomicAdd(p + 1, v.y);
        atomicAdd(p + 2, v.z);
        atomicAdd(p + 3, v.w);
    }
}

// Layer 2: 32 features, one lane per (edge, feature).
__global__ void scatter_edges_32(const float* __restrict__ x, float* __restrict__ acc,
                                 const int* __restrict__ src, const int* __restrict__ dst,
                                 long long E) {
    long long gid = (long long)blockIdx.x * blockDim.x + threadIdx.x;
    long long stride = (long long)gridDim.x * blockDim.x;
    for (long long w = gid; w < E * 32; w += stride) {
        long long e = w >> 5;
        int f = (int)(w & 31);
        atomicAdd(acc + (size_t)dst[e] * 32 + f, x[(size_t)src[e] * 32 + f]);
    }
}

// ----------------------------------------- GEMM (fp32 WMMA) + bias + BN stats
// t[N,64] = in[N,KIN] @ w[KIN,64] + b; accumulates per-column sum / sum-of-squares.
// Block = 128 threads (4 waves); each wave does one 16x16 tile (wave -> col tile).
template <int KIN>
__global__ __launch_bounds__(128)
void gemm16_bias_stats(const float* __restrict__ in, const float* __restrict__ w,
                       const float* __restrict__ b, float* __restrict__ out,
                       float* __restrict__ sum, float* __restrict__ sumsq, int N) {
    const int lane = threadIdx.x & 31;
    const int wv   = threadIdx.x >> 5;    // 0..3 -> 16-col tile
    const int l15  = lane & 15;
    const int hi   = lane >> 4;           // 0: lanes 0-15, 1: lanes 16-31
    const int row0 = blockIdx.x * 16;
    const int col0 = wv * 16;

    int arow = row0 + l15;
    if (arow > N - 1) arow = N - 1;       // clamp: keep EXEC full for WMMA
    const float* ap = in + (size_t)arow * KIN + hi * 2;

    v8f c = {};
#pragma unroll
    for (int k = 0; k < KIN; k += 4) {
        v2f a, bb;
        a.x = ap[k];          // A: K = k + 2*hi
        a.y = ap[k + 1];      //    K = k + 2*hi + 1
        bb.x = w[(size_t)(k + hi * 2)     * 64 + col0 + l15];
        bb.y = w[(size_t)(k + hi * 2 + 1) * 64 + col0 + l15];
        c = __builtin_amdgcn_wmma_f32_16x16x4_f32(false, a, false, bb,
                                                  (short)0, c, false, false);
    }

    const float bias = b[col0 + l15];
    float s = 0.0f, sq = 0.0f;
#pragma unroll
    for (int r = 0; r < 8; ++r) {
        const int m = r + hi * 8;         // C/D layout: vgpr r -> row r or r+8
        const float v = c[r] + bias;
        if (row0 + m < N) {
            out[(size_t)(row0 + m) * 64 + col0 + l15] = v;
            s += v; sq += v * v;
        }
    }
    atomicAdd(&sum[col0 + l15], s);
    atomicAdd(&sumsq[col0 + l15], sq);
}

// mu/var -> affine scale/shift for BN (64 channels, one block of 64 threads)
__global__ void bn_finalize(const float* __restrict__ sum, const float* __restrict__ sumsq,
                            const float* __restrict__ g, const float* __restrict__ bt,
                            float* __restrict__ scale, float* __restrict__ shift, float invN) {
    const int i = threadIdx.x;
    const float mu  = sum[i] * invN;
    const float var = sumsq[i] * invN - mu * mu;
    const float sc  = g[i] * rsqrtf(var + 1e-5f);
    scale[i] = sc;
    shift[i] = bt[i] - mu * sc;
}

// out[N,NOUT] = relu(t*scale+shift)[N,64] @ w[64,NOUT] + b   (fp32 WMMA, fused BN/ReLU)
template <int NOUT>
__global__ __launch_bounds__(32)
void bn_relu_gemm(const float* __restrict__ t, const float* __restrict__ scale,
                  const float* __restrict__ shift, const float* __restrict__ w,
                  const float* __restrict__ b, float* __restrict__ out, int N) {
    const int lane = threadIdx.x & 31;
    const int l15  = lane & 15;
    const int hi   = lane >> 4;
    const int row0 = blockIdx.x * 16;
    const int col0 = blockIdx.y * 16;

    int arow = row0 + l15;
    if (arow > N - 1) arow = N - 1;
    const float* ap = t + (size_t)arow * 64;

    v8f c = {};
#pragma unroll
    for (int k = 0; k < 64; k += 4) {
        const int k0 = k + hi * 2, k1 = k0 + 1;
        v2f a, bb;
        a.x = fmaxf(fmaf(ap[k0], scale[k0], shift[k0]), 0.0f);
        a.y = fmaxf(fmaf(ap[k1], scale[k1], shift[k1]), 0.0f);
        bb.x = w[(size_t)k0 * NOUT + col0 + l15];
        bb.y = w[(size_t)k1 * NOUT + col0 + l15];
        c = __builtin_amdgcn_wmma_f32_16x16x4_f32(false, a, false, bb,
                                                  (short)0, c, false, false);
    }

    const float bias = b[col0 + l15];
#pragma unroll
    for (int r = 0; r < 8; ++r) {
        const int m = r + hi * 8;
        if (row0 + m < N)
            out[(size_t)(row0 + m) * NOUT + col0 + l15] = c[r] + bias;
    }
}

// ------------------------------------------------ global mean pool (LDS-staged)
__global__ void pool_kernel(const float* __restrict__ h, const int* __restrict__ batch,
                            float* __restrict__ pooled, float* __restrict__ counts, int N) {
    __shared__ float sp[64 * 16];
    __shared__ float scnt[64];
    for (int i = threadIdx.x; i < 64 * 16; i += blockDim.x) sp[i] = 0.0f;
    for (int i = threadIdx.x; i < 64; i += blockDim.x) scnt[i] = 0.0f;
    __syncthreads();

    const int f = threadIdx.x & 15;
    int node = (int)((blockIdx.x * blockDim.x + threadIdx.x) >> 4);
    const int stride = (int)((gridDim.x * blockDim.x) >> 4);
    for (; node < N; node += stride) {
        const int g = batch[node];
        atomicAdd(&sp[g * 16 + f], h[(size_t)node * 16 + f]);
        if (f == 0) atomicAdd(&scnt[g], 1.0f);
    }
    __syncthreads();
    for (int i = threadIdx.x; i < 64 * 16; i += blockDim.x) atomicAdd(&pooled[i], sp[i]);
    for (int i = threadIdx.x; i < 64; i += blockDim.x) atomicAdd(&counts[i], scnt[i]);
}

__global__ void final_kernel(const float* __restrict__ pooled, const float* __restrict__ counts,
                             const float* __restrict__ lw, const float* __restrict__ lb,
                             float* __restrict__ out) {
    const int g = threadIdx.x;   // 64 graphs
    const float inv = 1.0f / fmaxf(counts[g], 1.0f);
    float acc = lb[0];
#pragma unroll
    for (int i = 0; i < 16; ++i) acc += pooled[g * 16 + i] * inv * lw[i];
    out[g] = 1.0f / (1.0f + __expf(-acc));
}

// ----------------------------------------------------------------- launcher
extern "C" void kernel_launch(void* const* d_in, const int* in_sizes, int n_in,
                              void* d_out, int out_size, void* d_ws, size_t ws_size,
                              hipStream_t stream) {
    const float* x    = (const float*)d_in[0];
    const int*  edge  = (const int*)d_in[1];   // jnp int64 -> int32 (no x64 in jax)
    const int*  batch = (const int*)d_in[2];
    const float* eps1 = (const float*)d_in[3];
    const float* eps2 = (const float*)d_in[4];
    const float* w1a = (const float*)d_in[5];  const float* b1a = (const float*)d_in[6];
    const float* g1  = (const float*)d_in[7];  const float* bt1 = (const float*)d_in[8];
    const float* w1b = (const float*)d_in[9];  const float* b1b = (const float*)d_in[10];
    const float* w2a = (const float*)d_in[11]; const float* b2a = (const float*)d_in[12];
    const float* g2  = (const float*)d_in[13]; const float* bt2 = (const float*)d_in[14];
    const float* w2b = (const float*)d_in[15]; const float* b2b = (const float*)d_in[16];
    const float* lw  = (const float*)d_in[17]; const float* lb  = (const float*)d_in[18];
    float* out = (float*)d_out;

    const long long N = in_sizes[0] / 128;
    const long long E = in_sizes[1] / 2;
    const int* src = edge;
    const int* dst = edge + E;

    // workspace layout (floats): A[N*128] | T[N*64] | H[N*32] | stats[1344]
    float* bufA  = (float*)d_ws;
    float* bufT  = bufA + (size_t)N * 128;
    float* bufH  = bufT + (size_t)N * 64;
    float* stats = bufH + (size_t)N * 32;
    float* sumv   = stats;
    float* sumsq  = stats + 64;
    float* scalev = stats + 128;
    float* shiftv = stats + 192;
    float* pooled = stats + 256;          // 64*16
    float* counts = stats + 256 + 1024;   // 64

    const int nTiles = (int)((N + 15) / 16);
    const float invN = 1.0f / (float)N;

    // 0) zero all small state (sum/sumsq/scale/shift/pooled/counts)
    zero_floats<<<(1344 + 255) / 256, 256, 0, stream>>>(stats, 1344);

    // ---- Layer 1 ----
    scale_copy4<<<2048, 256, 0, stream>>>(x, bufA, eps1, N * 128 / 4);
    scatter_edges_128<<<8192, 256, 0, stream>>>(x, bufA, src, dst, E);
    gemm16_bias_stats<128><<<nTiles, 128, 0, stream>>>(bufA, w1a, b1a, bufT, sumv, sumsq, (int)N);
    bn_finalize<<<1, 64, 0, stream>>>(sumv, sumsq, g1, bt1, scalev, shiftv, invN);
    bn_relu_gemm<32><<<dim3(nTiles, 2), 32, 0, stream>>>(bufT, scalev, shiftv, w1b, b1b, bufH, (int)N);

    // ---- Layer 2 ----
    zero_floats<<<1, 128, 0, stream>>>(stats, 128);  // re-zero sum/sumsq
    scale_copy4<<<2048, 256, 0, stream>>>(bufH, bufA, eps2, N * 32 / 4);
    scatter_edges_32<<<8192, 256, 0, stream>>>(bufH, bufA, src, dst, E);
    gemm16_bias_stats<32><<<nTiles, 128, 0, stream>>>(bufA, w2a, b2a, bufT, sumv, sumsq, (int)N);
    bn_finalize<<<1, 64, 0, stream>>>(sumv, sumsq, g2, bt2, scalev, shiftv, invN);
    bn_relu_gemm<16><<<dim3(nTiles, 1), 32, 0, stream>>>(bufT, scalev, shiftv, w2b, b2b, bufH, (int)N);

    // ---- Pool + classifier ----
    pool_kernel<<<400, 256, 0, stream>>>(bufH, batch, pooled, counts, (int)N);
    final_kernel<<<1, 64, 0, stream>>>(pooled, counts, lw, lb, out);
}